// LongTermStorage_87754771792455
// MI455X (gfx1250) — compile-verified
//
#include <hip/hip_runtime.h>
#include <hip/hip_bf16.h>

typedef __attribute__((ext_vector_type(2))) float v2f;
typedef __attribute__((ext_vector_type(8))) float v8f;

#define KEY_DIM   32
#define NB        4096     // histogram bins over [-1, 1]
#define CAP       4096     // max candidates kept for final sort
#define SORT_N    4096     // bitonic sort size (power of 2, >= CAP)
#define EPS       1e-8f
#define THRESHOLD 0.1f

// ---------------------------------------------------------------------------
// Kernel 0: zero histogram + candidate counter
// ---------------------------------------------------------------------------
__global__ void ltm_init_kernel(unsigned* hist, int* counter) {
    int i = blockIdx.x * blockDim.x + threadIdx.x;
    if (i < NB) hist[i] = 0u;
    if (i == 0) counter[0] = 0;
}

// ---------------------------------------------------------------------------
// Kernel 1: fused dot + norm via WMMA f32 16x16x4.
// One wave = 16 key rows. A = keys chunk (16x4), B = query replicated (4x16).
// Second accumulator with B := A gives the Gram matrix; diagonal = ||row||^2.
// ---------------------------------------------------------------------------
__global__ void __launch_bounds__(128)
ltm_sims_kernel(const float* __restrict__ query,
                const float* __restrict__ keys,
                float* __restrict__ sims, int N) {
    const int lane    = threadIdx.x & 31;
    const int wave    = threadIdx.x >> 5;
    const int waveId  = blockIdx.x * 4 + wave;
    const int rowBase = waveId * 16;
    if (rowBase >= N) return;                      // wave-uniform

    // query norm (uniform; tiny, stays in cache)
    float qs = 0.f;
    #pragma unroll
    for (int j = 0; j < KEY_DIM; ++j) { float q = query[j]; qs += q * q; }
    const float qn = fmaxf(sqrtf(qs), EPS);

    if (rowBase + 16 <= N) {                       // wave-uniform: full 16-row tile
        const int h = lane >> 4;                   // K-half select (0: K0,K1  1: K2,K3)
        const int m = lane & 15;                   // row within tile

        // Query fragments: B[k][n] = query[k] for all n (all columns identical)
        v2f qfrag[8];
        #pragma unroll
        for (int c = 0; c < 8; ++c)
            qfrag[c] = *reinterpret_cast<const v2f*>(query + 4 * c + 2 * h);

        const float* rowPtr = keys + (size_t)(rowBase + m) * KEY_DIM + 2 * h;

        v8f accDot  = {};
        v8f accGram = {};
        #pragma unroll
        for (int c = 0; c < 8; ++c) {
            v2f a = *reinterpret_cast<const v2f*>(rowPtr + 4 * c);
            accDot  = __builtin_amdgcn_wmma_f32_16x16x4_f32(
                          false, a, false, qfrag[c], (short)0, accDot,  false, false);
            accGram = __builtin_amdgcn_wmma_f32_16x16x4_f32(
                          false, a, false, a,        (short)0, accGram, false, false);
        }

        // C/D layout: VGPR r -> lanes 0-15: D[r][lane]; lanes 16-31: D[8+r][lane-16]
        // Row r     : dot & norm^2 at lane r,    VGPR r (diagonal D[r][r])
        // Row 8+r   : dot & norm^2 at lane 24+r, VGPR r (diagonal D[8+r][8+r])
        #pragma unroll
        for (int r = 0; r < 8; ++r) {
            if (lane == r) {
                float kn = fmaxf(sqrtf(accGram[r]), EPS);
                sims[rowBase + r] = accDot[r] / (kn * qn);
            }
            if (lane == 24 + r) {
                float kn = fmaxf(sqrtf(accGram[r]), EPS);
                sims[rowBase + 8 + r] = accDot[r] / (kn * qn);
            }
        }
    } else {
        // scalar tail (never taken for N = 1M, kept for generality)
        int row = rowBase + lane;
        if (lane < 16 && row < N) {
            float dot = 0.f, n2 = 0.f;
            for (int j = 0; j < KEY_DIM; ++j) {
                float kv = keys[(size_t)row * KEY_DIM + j];
                dot += kv * query[j];
                n2  += kv * kv;
            }
            sims[row] = dot / (fmaxf(sqrtf(n2), EPS) * qn);
        }
    }
}

// ---------------------------------------------------------------------------
// Kernel 2: histogram of sims into NB bins over [-1, 1]
// ---------------------------------------------------------------------------
__global__ void __launch_bounds__(256)
ltm_hist_kernel(const float* __restrict__ sims, int N, unsigned* __restrict__ hist) {
    __shared__ unsigned lhist[NB];
    for (int i = threadIdx.x; i < NB; i += blockDim.x) lhist[i] = 0u;
    __syncthreads();

    int stride = gridDim.x * blockDim.x;
    for (int i = blockIdx.x * blockDim.x + threadIdx.x; i < N; i += stride) {
        float s = sims[i];
        int b = (int)((s * 0.5f + 0.5f) * (float)NB);
        b = b < 0 ? 0 : (b >= NB ? NB - 1 : b);
        atomicAdd(&lhist[b], 1u);
    }
    __syncthreads();
    for (int i = threadIdx.x; i < NB; i += blockDim.x)
        if (lhist[i]) atomicAdd(&hist[i], lhist[i]);
}

// ---------------------------------------------------------------------------
// Kernel 3: scan histogram from the top; emit a conservative threshold whose
// candidate set provably contains the true top-k.
// ---------------------------------------------------------------------------
__global__ void ltm_scan_kernel(const unsigned* __restrict__ hist, float* thrOut, int k) {
    if (threadIdx.x != 0 || blockIdx.x != 0) return;
    unsigned cum = 0;
    int b = 0;
    for (int i = NB - 1; i >= 0; --i) {
        cum += hist[i];
        if (cum >= (unsigned)k) { b = i; break; }
    }
    const float binw = 2.0f / (float)NB;
    // lower edge of bin b, minus one extra bin of float-rounding margin
    *thrOut = -1.0f + (float)b * binw - binw;
}

// ---------------------------------------------------------------------------
// Kernel 4: compact candidates above threshold
// ---------------------------------------------------------------------------
__global__ void __launch_bounds__(256)
ltm_compact_kernel(const float* __restrict__ sims, int N,
                   const float* __restrict__ thr, int* __restrict__ counter,
                   float* __restrict__ candSim, int* __restrict__ candIdx) {
    float t = *thr;
    int stride = gridDim.x * blockDim.x;
    for (int i = blockIdx.x * blockDim.x + threadIdx.x; i < N; i += stride) {
        float s = sims[i];
        if (s >= t) {
            int p = atomicAdd(counter, 1);
            if (p < CAP) { candSim[p] = s; candIdx[p] = i; }
        }
    }
}

// ---------------------------------------------------------------------------
// Kernel 5: single-block bitonic sort (sim desc, idx asc) + masked gather.
// Order is deterministic regardless of atomic arrival order in kernel 4.
// ---------------------------------------------------------------------------
__global__ void __launch_bounds__(1024)
ltm_topk_kernel(const float* __restrict__ candSim, const int* __restrict__ candIdx,
                const int* __restrict__ counter, const float* __restrict__ values,
                float* __restrict__ out, int k, int DIM) {
    __shared__ float sSim[SORT_N];
    __shared__ int   sIdx[SORT_N];

    int count = *counter;
    if (count > CAP) count = CAP;

    for (int i = threadIdx.x; i < SORT_N; i += blockDim.x) {
        if (i < count) { sSim[i] = candSim[i]; sIdx[i] = candIdx[i]; }
        else           { sSim[i] = -__builtin_inff(); sIdx[i] = 0x7fffffff; }
    }
    __syncthreads();

    // bitonic sort; "before(a,b)" == a precedes b in final order
    for (int size = 2; size <= SORT_N; size <<= 1) {
        for (int stride = size >> 1; stride > 0; stride >>= 1) {
            for (int i = threadIdx.x; i < SORT_N; i += blockDim.x) {
                int j = i ^ stride;
                if (j > i) {
                    float si = sSim[i], sj = sSim[j];
                    int   ii = sIdx[i], ij = sIdx[j];
                    bool before = (si > sj) || (si == sj && ii < ij);
                    bool up = ((i & size) == 0);
                    if (up ? !before : before) {
                        sSim[i] = sj; sSim[j] = si;
                        sIdx[i] = ij; sIdx[j] = ii;
                    }
                }
            }
            __syncthreads();
        }
    }

    // out layout: [k*DIM values][k sims]
    for (int i = threadIdx.x; i < k; i += blockDim.x) {
        float s = sSim[i];
        bool m = (i < count) && (s >= THRESHOLD);
        out[(size_t)k * DIM + i] = m ? s : 0.0f;
    }
    int total = k * DIM;
    for (int e = threadIdx.x; e < total; e += blockDim.x) {
        int i = e / DIM, j = e - i * DIM;
        float s = sSim[i];
        bool m = (i < count) && (s >= THRESHOLD);
        out[e] = m ? values[(size_t)sIdx[i] * DIM + j] : 0.0f;
    }
}

// ---------------------------------------------------------------------------
extern "C" void kernel_launch(void* const* d_in, const int* in_sizes, int n_in,
                              void* d_out, int out_size, void* d_ws, size_t ws_size,
                              hipStream_t stream) {
    const float* query  = (const float*)d_in[0];
    const float* keys   = (const float*)d_in[1];
    const float* values = (const float*)d_in[2];

    const int keyDim = in_sizes[0];                 // 32
    const int N      = in_sizes[1] / keyDim;        // 1,000,000
    const int DIM    = in_sizes[2] / N;             // 512
    const int k      = out_size / (DIM + 1);        // 64
    (void)n_in; (void)ws_size; (void)keyDim;

    // workspace layout (all 4-byte aligned)
    float*    sims    = (float*)d_ws;               // N floats
    unsigned* hist    = (unsigned*)(sims + N);      // NB
    int*      counter = (int*)(hist + NB);          // 1 (+1 pad implied below)
    float*    thr     = (float*)(counter + 1);      // 1
    float*    candSim = (float*)(counter + 2);      // CAP
    int*      candIdx = (int*)(candSim + CAP);      // CAP

    ltm_init_kernel<<<(NB + 255) / 256, 256, 0, stream>>>(hist, counter);

    int waves   = (N + 15) / 16;
    int blocks1 = (waves + 3) / 4;                  // 4 waves (128 threads) per block
    ltm_sims_kernel<<<blocks1, 128, 0, stream>>>(query, keys, sims, N);

    ltm_hist_kernel<<<1024, 256, 0, stream>>>(sims, N, hist);
    ltm_scan_kernel<<<1, 32, 0, stream>>>(hist, thr, k);
    ltm_compact_kernel<<<1024, 256, 0, stream>>>(sims, N, thr, counter, candSim, candIdx);
    ltm_topk_kernel<<<1, 1024, 0, stream>>>(candSim, candIdx, counter, values,
                                            (float*)d_out, k, DIM);
}